// RecurrentNNStandardTorch_55740085567662
// MI455X (gfx1250) — compile-verified
//
#include <hip/hip_runtime.h>
#include <math.h>

typedef __attribute__((ext_vector_type(2))) float v2f;
typedef __attribute__((ext_vector_type(4))) float v4f;
typedef __attribute__((ext_vector_type(8))) float v8f;

constexpr int kN = 4096;           // samples
constexpr int kS = 50;             // seq len per sample
constexpr int kF = 32;             // input features
constexpr int kH = 64;             // hidden
constexpr int kG = 3 * kH;         // 192 gate rows
constexpr int kT = kN * kS;        // 204800 total timesteps (one chained scan)

// ---------------------------------------------------------------------------
// Phase 1: xp[T,192] = x[T,32] @ W_ih^T + b_ih  via V_WMMA_F32_16X16X4_F32.
// One wave per 16-row strip of x. A-tile (16x32 f32) loaded once into 8 v2f
// regs per lane following the ISA f32 A layout:
//   lane l<16 : row = l,      a[s] = {x[row][4s+0], x[row][4s+1]}
//   lane l>=16: row = l-16,   a[s] = {x[row][4s+2], x[row][4s+3]}
// B (4x16 per step) mirrors this: VGPR0 holds K=k0 (lanes 0-15) / K=k0+2
// (lanes 16-31), VGPR1 holds K=k0+1 / K=k0+3, N = lane%16.
// D layout: VGPR v -> rows v (lanes 0-15) and v+8 (lanes 16-31), N = lane%16.
// ---------------------------------------------------------------------------
__global__ __launch_bounds__(32) void gru_xproj_wmma(
    const float* __restrict__ x, const float* __restrict__ W_ih,
    const float* __restrict__ b_ih, float* __restrict__ xp)
{
  const int lane = threadIdx.x;
  const int l16  = lane & 15;
  const int half = lane >> 4;            // 0: K pair {0,1}, 1: K pair {2,3}
  const int row0 = blockIdx.x * 16;

  const float* xr = x + (size_t)(row0 + l16) * kF;
  v2f a[8];
#pragma unroll
  for (int s = 0; s < 8; ++s)
    a[s] = *(const v2f*)(xr + 4 * s + 2 * half);

#pragma unroll
  for (int ct = 0; ct < 12; ++ct) {      // 12 column tiles of 16 (192 gates)
    const int col   = ct * 16 + l16;
    const float* wr = W_ih + (size_t)col * kF;   // B[k][col] = W_ih[col][k]
    v8f c = {0.f, 0.f, 0.f, 0.f, 0.f, 0.f, 0.f, 0.f};
#pragma unroll
    for (int s = 0; s < 8; ++s) {        // K = 32 in 8 steps of 4
      v2f b = *(const v2f*)(wr + 4 * s + 2 * half);
      c = __builtin_amdgcn_wmma_f32_16x16x4_f32(
          /*neg_a=*/false, a[s], /*neg_b=*/false, b,
          /*c_mod=*/(short)0, c, /*reuse_a=*/false, /*reuse_b=*/false);
    }
    const float bias = b_ih[col];
    float* outp = xp + (size_t)(row0 + half * 8) * kG + col;
#pragma unroll
    for (int v = 0; v < 8; ++v)
      outp[(size_t)v * kG] = c[v] + bias;
  }
}

// ---------------------------------------------------------------------------
// Phase 2: the 204800-step sequential GRU scan. 1 block, 192 threads (6
// waves). Thread j owns W_hh row j entirely in VGPRs (64 floats); h/r/z live
// in LDS (same-address reads broadcast). Two barriers per step. FC head
// (O=1) folded in at each sample's last step.
// ---------------------------------------------------------------------------
__global__ __launch_bounds__(192) void gru_scan(
    const float* __restrict__ xp, const float* __restrict__ W_hh,
    const float* __restrict__ b_hh, const float* __restrict__ W_out,
    const float* __restrict__ b_out, float* __restrict__ out)
{
  __shared__ float h_lds[kH];
  __shared__ float r_lds[kH];
  __shared__ float z_lds[kH];
  __shared__ float po_lds[kH];

  const int j = threadIdx.x;             // gate row 0..191
  const int k = j & 63;                  // hidden index within gate

  float w[kH];                           // W_hh row j resident in VGPRs
#pragma unroll
  for (int q = 0; q < kH; ++q) w[q] = W_hh[(size_t)j * kH + q];
  const float bh = b_hh[j];
  const float wo = W_out[k];             // used by threads 128..191 only

  if (j < kH) h_lds[j] = 0.0f;           // h0 = 0
  __syncthreads();

  const v4f* h4 = (const v4f*)h_lds;
  float xcur = xp[j];                    // software-pipelined xp stream

  for (int t = 0; t < kT; ++t) {
    float xnext = 0.0f;
    if (t + 1 < kT) xnext = xp[(size_t)(t + 1) * kG + j];
    if (t + 8 < kT) __builtin_prefetch(&xp[(size_t)(t + 8) * kG + j], 0, 0);

    // gh_j = W_hh[j] . h + b_hh[j]  (4 chains to break FMA latency)
    float g0 = 0.f, g1 = 0.f, g2 = 0.f, g3 = 0.f;
#pragma unroll
    for (int q = 0; q < kH / 4; ++q) {
      v4f hv = h4[q];
      g0 = fmaf(w[4 * q + 0], hv.x, g0);
      g1 = fmaf(w[4 * q + 1], hv.y, g1);
      g2 = fmaf(w[4 * q + 2], hv.z, g2);
      g3 = fmaf(w[4 * q + 3], hv.w, g3);
    }
    const float gh  = bh + ((g0 + g1) + (g2 + g3));
    const float pre = xcur + gh;

    if (j < 64) {
      r_lds[j] = 1.0f / (1.0f + __expf(-pre));          // reset gate
    } else if (j < 128) {
      z_lds[j - 64] = 1.0f / (1.0f + __expf(-pre));     // update gate
    }
    __syncthreads();   // r,z visible; all dot-reads of h done

    const bool last = ((t + 1) % kS) == 0;
    if (j >= 128) {
      const float r  = r_lds[k];
      const float z  = z_lds[k];
      const float hp = h_lds[k];
      const float ng = tanhf(xcur + r * gh);            // candidate
      const float hn = (1.0f - z) * ng + z * hp;        // PyTorch update
      h_lds[k] = hn;
      if (last) po_lds[k] = hn * wo;                    // FC head partials
    }
    __syncthreads();   // h_new (and po) visible

    if (last && j == 0) {
      float s = b_out[0];
      for (int q = 0; q < kH; ++q) s += po_lds[q];
      out[t / kS] = s;
    }
    xcur = xnext;
  }
}

// ---------------------------------------------------------------------------
extern "C" void kernel_launch(void* const* d_in, const int* in_sizes, int n_in,
                              void* d_out, int out_size, void* d_ws, size_t ws_size,
                              hipStream_t stream) {
  const float* x     = (const float*)d_in[0];
  const float* W_ih  = (const float*)d_in[1];
  const float* W_hh  = (const float*)d_in[2];
  const float* b_ih  = (const float*)d_in[3];
  const float* b_hh  = (const float*)d_in[4];
  const float* W_out = (const float*)d_in[5];
  const float* b_out = (const float*)d_in[6];
  float* out = (float*)d_out;
  float* xp  = (float*)d_ws;             // needs T*192*4 = 157,286,400 bytes

  gru_xproj_wmma<<<kT / 16, 32, 0, stream>>>(x, W_ih, b_ih, xp);
  gru_scan<<<1, 192, 0, stream>>>(xp, W_hh, b_hh, W_out, b_out, out);
}